// HGNN_8134668059208
// MI455X (gfx1250) — compile-verified
//
#include <hip/hip_runtime.h>
#include <math.h>

// ---------------------------------------------------------------------------
// Types for CDNA5 WMMA f32 16x16x4:
//   A 16x4 f32  -> 2 VGPRs/lane -> v2f
//   B 4x16 f32  -> 2 VGPRs/lane -> v2f
//   C/D 16x16   -> 8 VGPRs/lane -> v8f
// ---------------------------------------------------------------------------
typedef __attribute__((ext_vector_type(2))) float v2f;
typedef __attribute__((ext_vector_type(8))) float v8f;

#define BM 64
#define BN 128
#define BK 32
#define NT 128     // threads per block (4 waves: 2 along M x 2 along N)
#define ASTR 80    // padded LDS stride for A tile (2*80  % 64 == 32 -> half-waves on disjoint banks)
#define BSTR 144   // padded LDS stride for B tile (2*144 % 64 == 32)

// ---------------------------------------------------------------------------
// Generic fp32 WMMA GEMM: C[M,N] = op(A) @ op(B), optional row-scale on B's
// K dimension (scale[k]) and fused ReLU on output.
//   transA: logical A[m][k] = A[k*lda + m]   (used for H^T)
//   transB: logical B[k][n] = B[n*ldb + k]   (used for W^T)
// Requires M % 64 == 0, K % 32 == 0; N is guarded (N=40 case).
// Each wave computes a 32x64 patch = 2x4 16x16 fragments:
//   8 WMMA per 12 LDS dwords per K-slice (A frags reused 4x, B frags 2x).
// ---------------------------------------------------------------------------
__launch_bounds__(NT)
__global__ void wmma_gemm_f32(const float* __restrict__ A,
                              const float* __restrict__ B,
                              float* __restrict__ C,
                              int M, int N, int K,
                              int lda, int ldb, int ldc,
                              int transA, int transB, int doRelu,
                              const float* __restrict__ rowScaleB)
{
    __shared__ float As[BK * ASTR];
    __shared__ float Bs[BK * BSTR];

    const int tid  = threadIdx.x;
    const int m0   = blockIdx.y * BM;
    const int n0   = blockIdx.x * BN;

    const int wave  = tid >> 5;
    const int lane  = tid & 31;
    const int waveM = wave & 1;    // 2 waves along M (32 rows each)
    const int waveN = wave >> 1;   // 2 waves along N (64 cols each)
    const int lr    = lane & 15;
    const int lh    = lane >> 4;

    const v8f z8 = {0.f,0.f,0.f,0.f,0.f,0.f,0.f,0.f};
    v8f acc[2][4];
    #pragma unroll
    for (int mi = 0; mi < 2; ++mi)
        #pragma unroll
        for (int ni = 0; ni < 4; ++ni) acc[mi][ni] = z8;

    for (int k0 = 0; k0 < K; k0 += BK) {
        __syncthreads();
        // ---- stage A tile into As[k][m] (coalesced global reads) ----
        if (!transA) {
            #pragma unroll
            for (int i = tid; i < BM * BK; i += NT) {
                int m = i >> 5, k = i & 31;             // k fastest: contiguous
                As[k * ASTR + m] = A[(size_t)(m0 + m) * lda + (k0 + k)];
            }
        } else {
            #pragma unroll
            for (int i = tid; i < BM * BK; i += NT) {
                int m = i & (BM - 1), k = i >> 6;       // m fastest: contiguous
                As[k * ASTR + m] = A[(size_t)(k0 + k) * lda + (m0 + m)];
            }
        }
        // ---- stage B tile into Bs[k][n] (with optional per-k scale) ----
        if (!transB) {
            #pragma unroll
            for (int i = tid; i < BK * BN; i += NT) {
                int k = i >> 7, n = i & (BN - 1);       // n fastest: contiguous
                float val = (n0 + n < N) ? B[(size_t)(k0 + k) * ldb + (n0 + n)] : 0.f;
                if (rowScaleB) val *= rowScaleB[k0 + k];
                Bs[k * BSTR + n] = val;
            }
        } else {
            #pragma unroll
            for (int i = tid; i < BK * BN; i += NT) {
                int k = i & 31, n = i >> 5;             // k fastest: contiguous
                float val = (n0 + n < N) ? B[(size_t)(n0 + n) * ldb + (k0 + k)] : 0.f;
                if (rowScaleB) val *= rowScaleB[k0 + k];
                Bs[k * BSTR + n] = val;
            }
        }
        // L2 prefetch of next K tile (global_prefetch_b8 on gfx1250)
        if (k0 + BK < K) {
            int kn = k0 + BK;
            const float* pa = transA ? (A + (size_t)kn * lda + m0 + tid)
                                     : (A + (size_t)(m0 + (tid >> 1)) * lda + kn);
            __builtin_prefetch(pa, 0, 1);
        }
        __syncthreads();

        // ---- 8 K-slices of 4; 2x4 WMMA tiles per wave ----
        #pragma unroll
        for (int kk = 0; kk < BK; kk += 4) {
            const int ka = kk + 2 * lh;
            const int mA = waveM * 32 + lr;
            const int nB = waveN * 64 + lr;
            v2f a0, a1, b0, b1, b2, b3;
            a0[0] = As[ka * ASTR + mA];        a0[1] = As[(ka + 1) * ASTR + mA];
            a1[0] = As[ka * ASTR + mA + 16];   a1[1] = As[(ka + 1) * ASTR + mA + 16];
            b0[0] = Bs[ka * BSTR + nB];        b0[1] = Bs[(ka + 1) * BSTR + nB];
            b1[0] = Bs[ka * BSTR + nB + 16];   b1[1] = Bs[(ka + 1) * BSTR + nB + 16];
            b2[0] = Bs[ka * BSTR + nB + 32];   b2[1] = Bs[(ka + 1) * BSTR + nB + 32];
            b3[0] = Bs[ka * BSTR + nB + 48];   b3[1] = Bs[(ka + 1) * BSTR + nB + 48];
            acc[0][0] = __builtin_amdgcn_wmma_f32_16x16x4_f32(false, a0, false, b0,
                          (short)0, acc[0][0], false, false);
            acc[0][1] = __builtin_amdgcn_wmma_f32_16x16x4_f32(false, a0, false, b1,
                          (short)0, acc[0][1], false, false);
            acc[0][2] = __builtin_amdgcn_wmma_f32_16x16x4_f32(false, a0, false, b2,
                          (short)0, acc[0][2], false, false);
            acc[0][3] = __builtin_amdgcn_wmma_f32_16x16x4_f32(false, a0, false, b3,
                          (short)0, acc[0][3], false, false);
            acc[1][0] = __builtin_amdgcn_wmma_f32_16x16x4_f32(false, a1, false, b0,
                          (short)0, acc[1][0], false, false);
            acc[1][1] = __builtin_amdgcn_wmma_f32_16x16x4_f32(false, a1, false, b1,
                          (short)0, acc[1][1], false, false);
            acc[1][2] = __builtin_amdgcn_wmma_f32_16x16x4_f32(false, a1, false, b2,
                          (short)0, acc[1][2], false, false);
            acc[1][3] = __builtin_amdgcn_wmma_f32_16x16x4_f32(false, a1, false, b3,
                          (short)0, acc[1][3], false, false);
        }
    }

    // ---- store: D layout lane lr = N, vgpr r -> M = 8*lh + r ----
    #pragma unroll
    for (int mi = 0; mi < 2; ++mi)
        #pragma unroll
        for (int ni = 0; ni < 4; ++ni)
            #pragma unroll
            for (int r = 0; r < 8; ++r) {
                int m = m0 + waveM * 32 + mi * 16 + lh * 8 + r;
                int n = n0 + waveN * 64 + ni * 16 + lr;
                if (n < N) {
                    float v = acc[mi][ni][r];
                    if (doRelu) v = fmaxf(v, 0.f);
                    C[(size_t)m * ldc + n] = v;
                }
            }
}

// ---------------------------------------------------------------------------
// d_v (vertex degrees), rv = 1/sqrt(d_v+eps), d_e (edge degrees). grid=12.
// ---------------------------------------------------------------------------
__global__ void degrees_kernel(const float* __restrict__ H,
                               float* dv, float* rv, float* de)
{
    int b = blockIdx.x, t = threadIdx.x;
    if (b < 8) {
        int row = b * 256 + t;
        const float* hr = H + (size_t)row * 1024;
        float s = 0.f;
        for (int e = 0; e < 1024; ++e) s += hr[e];
        dv[row] = s;
        rv[row] = rsqrtf(s + 1e-10f);
    } else {
        int e = (b - 8) * 256 + t;
        float s = 0.f;
        for (int k = 0; k < 2048; ++k) s += H[(size_t)k * 1024 + e];
        de[e] = s;
    }
}

// r[k] = (sum_f X[k][f]^2) / (d_v[k]+eps).  grid = 2048 blocks.
__global__ void rowstats_kernel(const float* __restrict__ X,
                                const float* __restrict__ dv,
                                float* __restrict__ r, int F)
{
    __shared__ float red[256];
    int row = blockIdx.x, t = threadIdx.x;
    const float* xr = X + (size_t)row * F;
    float s = 0.f;
    for (int f = t; f < F; f += 256) { float v = xr[f]; s += v * v; }
    red[t] = s; __syncthreads();
    for (int o = 128; o > 0; o >>= 1) { if (t < o) red[t] += red[t + o]; __syncthreads(); }
    if (t == 0) r[row] = red[0] / (dv[row] + 1e-10f);
}

// q[e] = sum_k H[k][e] * r[k].  grid = 4 blocks.
__global__ void qvec_kernel(const float* __restrict__ H,
                            const float* __restrict__ r, float* q)
{
    int e = blockIdx.x * 256 + threadIdx.x;
    float s = 0.f;
    for (int k = 0; k < 2048; ++k) s += H[(size_t)k * 1024 + e] * r[k];
    q[e] = s;
}

// perE[e] = (d_e*q[e] - ||m_e||^2) / (d_e+eps).  grid = 1024 blocks.
__global__ void edge_energy_kernel(const float* __restrict__ Mm,
                                   const float* __restrict__ q,
                                   const float* __restrict__ de,
                                   float* perE, int F)
{
    __shared__ float red[256];
    int e = blockIdx.x, t = threadIdx.x;
    const float* mr = Mm + (size_t)e * F;
    float s = 0.f;
    for (int f = t; f < F; f += 256) { float v = mr[f]; s += v * v; }
    red[t] = s; __syncthreads();
    for (int o = 128; o > 0; o >>= 1) { if (t < o) red[t] += red[t + o]; __syncthreads(); }
    if (t == 0) { float d = de[e]; perE[e] = (d * q[e] - red[0]) / (d + 1e-10f); }
}

// partial[b] = partial dot(a,b) over grid-stride range (deterministic).
__global__ void dot_partial_kernel(const float* __restrict__ a,
                                   const float* __restrict__ b,
                                   int n, float* partial)
{
    __shared__ float red[256];
    int t = threadIdx.x;
    float s = 0.f;
    for (int i = blockIdx.x * 256 + t; i < n; i += gridDim.x * 256) s += a[i] * b[i];
    red[t] = s; __syncthreads();
    for (int o = 128; o > 0; o >>= 1) { if (t < o) red[t] += red[t + o]; __syncthreads(); }
    if (t == 0) partial[blockIdx.x] = red[0];
}

// dst[0] = sum(src[0..n)).  Single block.
__global__ void final_sum_kernel(const float* __restrict__ src, int n, float* dst)
{
    __shared__ float red[256];
    int t = threadIdx.x;
    float s = 0.f;
    for (int i = t; i < n; i += 256) s += src[i];
    red[t] = s; __syncthreads();
    for (int o = 128; o > 0; o >>= 1) { if (t < o) red[t] += red[t + o]; __syncthreads(); }
    if (t == 0) dst[0] = red[0];
}

// Spectral norm of each weight via power iteration on W^T W. grid = 10 blocks.
__global__ void __launch_bounds__(512)
spectral_norm_kernel(const float* __restrict__ Wf,
                     const float* __restrict__ Wm,
                     const float* __restrict__ Wl, float* out)
{
    __shared__ float v[512];
    __shared__ float u[512];
    __shared__ float red[512];
    int layer = blockIdx.x, t = threadIdx.x;
    const float* W; int R;
    if (layer == 0)       { W = Wf; R = 512; }
    else if (layer == 9)  { W = Wl; R = 40;  }
    else                  { W = Wm + (size_t)(layer - 1) * 512 * 512; R = 512; }

    v[t] = 1.f + 0.0009765625f * (float)t;
    __syncthreads();
    for (int it = 0; it < 50; ++it) {
        float s = 0.f;
        if (t < R) {
            const float* row = W + (size_t)t * 512;
            for (int c = 0; c < 512; ++c) s += row[c] * v[c];
        }
        u[t] = (t < R) ? s : 0.f;
        __syncthreads();
        float s2 = 0.f;
        for (int rI = 0; rI < R; ++rI) s2 += W[(size_t)rI * 512 + t] * u[rI];
        red[t] = s2 * s2; __syncthreads();
        for (int o = 256; o > 0; o >>= 1) { if (t < o) red[t] += red[t + o]; __syncthreads(); }
        float nrm = red[0];
        v[t] = s2 * rsqrtf(nrm + 1e-30f);
        __syncthreads();
    }
    float s = 0.f;
    if (t < R) {
        const float* row = W + (size_t)t * 512;
        for (int c = 0; c < 512; ++c) s += row[c] * v[c];
    }
    red[t] = s * s; __syncthreads();
    for (int o = 256; o > 0; o >>= 1) { if (t < o) red[t] += red[t + o]; __syncthreads(); }
    if (t == 0) out[layer] = sqrtf(red[0]);
}

// y = A @ x, block per row.
__global__ void matvec_kernel(const float* __restrict__ A,
                              const float* __restrict__ x,
                              float* __restrict__ y, int n)
{
    __shared__ float red[256];
    int row = blockIdx.x, t = threadIdx.x;
    const float* ar = A + (size_t)row * n;
    float s = 0.f;
    for (int k = t; k < n; k += 256) s += ar[k] * x[k];
    red[t] = s; __syncthreads();
    for (int o = 128; o > 0; o >>= 1) { if (t < o) red[t] += red[t + o]; __syncthreads(); }
    if (t == 0) y[row] = red[0];
}

__global__ void sg_init_kernel(float* v, int n)
{
    int i = blockIdx.x * 256 + threadIdx.x;
    if (i < n) v[i] = 1.f + 0.001f * (float)(i & 63);
}

// v <- normalize(c*v - u).  Single block.
__global__ void sg_update_kernel(float* __restrict__ v,
                                 const float* __restrict__ u, int n, float c)
{
    __shared__ float red[256];
    int t = threadIdx.x;
    float s = 0.f;
    for (int i = t; i < n; i += 256) { float w = c * v[i] - u[i]; s += w * w; }
    red[t] = s; __syncthreads();
    for (int o = 128; o > 0; o >>= 1) { if (t < o) red[t] += red[t + o]; __syncthreads(); }
    float inv = rsqrtf(red[0] + 1e-30f);
    for (int i = t; i < n; i += 256) v[i] = (c * v[i] - u[i]) * inv;
}

// out[0] = sqrt(max(v . u, 0))  with u = L(Lv), v unit -> |lambda|_min estimate.
__global__ void sg_final_kernel(const float* __restrict__ v,
                                const float* __restrict__ u, int n, float* out)
{
    __shared__ float red[256];
    int t = threadIdx.x;
    float s = 0.f;
    for (int i = t; i < n; i += 256) s += v[i] * u[i];
    red[t] = s; __syncthreads();
    for (int o = 128; o > 0; o >>= 1) { if (t < o) red[t] += red[t + o]; __syncthreads(); }
    if (t == 0) out[0] = sqrtf(fmaxf(red[0], 0.f));
}

// log_softmax over the N (row) dimension, one block per class column.
__global__ void logsoftmax_col_kernel(const float* __restrict__ X,
                                      float* __restrict__ out, int n, int C)
{
    __shared__ float red[256];
    int col = blockIdx.x, t = threadIdx.x;
    float mx = -3.4e38f;
    for (int i = t; i < n; i += 256) mx = fmaxf(mx, X[(size_t)i * C + col]);
    red[t] = mx; __syncthreads();
    for (int o = 128; o > 0; o >>= 1) { if (t < o) red[t] = fmaxf(red[t], red[t + o]); __syncthreads(); }
    mx = red[0]; __syncthreads();
    float s = 0.f;
    for (int i = t; i < n; i += 256) s += expf(X[(size_t)i * C + col] - mx);
    red[t] = s; __syncthreads();
    for (int o = 128; o > 0; o >>= 1) { if (t < o) red[t] += red[t + o]; __syncthreads(); }
    float lse = mx + logf(red[0]);
    for (int i = t; i < n; i += 256)
        out[(size_t)i * C + col] = X[(size_t)i * C + col] - lse;
}

// ---------------------------------------------------------------------------
extern "C" void kernel_launch(void* const* d_in, const int* in_sizes, int n_in,
                              void* d_out, int out_size, void* d_ws, size_t ws_size,
                              hipStream_t stream)
{
    (void)in_sizes; (void)n_in; (void)out_size; (void)ws_size;
    const float* X0 = (const float*)d_in[0];   // [1,2048,512]
    const float* H  = (const float*)d_in[1];   // [2048,1024]
    const float* L  = (const float*)d_in[2];   // [1,2048,2048]
    const float* Wf = (const float*)d_in[3];   // [512,512]
    const float* Wm = (const float*)d_in[4];   // [8,512,512]
    const float* Wl = (const float*)d_in[5];   // [40,512]
    float* out = (float*)d_out;

    float* ws   = (float*)d_ws;
    float* T    = ws;  ws += 2048 * 512;
    float* Xa   = ws;  ws += 2048 * 512;
    float* Xb   = ws;  ws += 2048 * 512;
    float* U    = ws;  ws += 2048 * 512;
    float* Mm   = ws;  ws += 1024 * 512;
    float* dv   = ws;  ws += 2048;
    float* rv   = ws;  ws += 2048;
    float* de   = ws;  ws += 1024;
    float* rvec = ws;  ws += 2048;
    float* qv   = ws;  ws += 1024;
    float* perE = ws;  ws += 1024;
    float* part = ws;  ws += 256;
    float* sgv  = ws;  ws += 2048;
    float* sgt  = ws;  ws += 2048;
    float* sgu  = ws;  ws += 2048;

    float* out_logp = out;            // 2048*40
    float* out_en   = out + 81920;    // 10
    float* out_enx  = out + 81930;    // 10
    float* out_sn   = out + 81940;    // 10
    float* out_gap  = out + 81950;    // 1

    degrees_kernel<<<12, 256, 0, stream>>>(H, dv, rv, de);

    const float* Xcur = X0;
    float* bufs[2] = {Xa, Xb};
    for (int layer = 0; layer < 10; ++layer) {
        int outN = (layer == 9) ? 40 : 512;
        const float* W = (layer == 0) ? Wf
                        : ((layer == 9) ? Wl : Wm + (size_t)(layer - 1) * 512 * 512);
        float* Xn = bufs[layer & 1];

        // T = L @ Xcur                    [2048,512]
        wmma_gemm_f32<<<dim3(512 / BN, 2048 / BM), NT, 0, stream>>>(
            L, Xcur, T, 2048, 512, 2048, 2048, 512, 512, 0, 0, 0, nullptr);
        // Xn = relu(T @ W^T)              [2048,outN]
        wmma_gemm_f32<<<dim3((outN + BN - 1) / BN, 2048 / BM), NT, 0, stream>>>(
            T, W, Xn, 2048, outN, 512, 512, 512, outN, 0, 1, 1, nullptr);
        // U = L @ Xn (for trace energy)   [2048,outN]
        wmma_gemm_f32<<<dim3((outN + BN - 1) / BN, 2048 / BM), NT, 0, stream>>>(
            L, Xn, U, 2048, outN, 2048, 2048, outN, outN, 0, 0, 0, nullptr);
        // energy = sum(Xn .* U)
        dot_partial_kernel<<<256, 256, 0, stream>>>(Xn, U, 2048 * outN, part);
        final_sum_kernel<<<1, 256, 0, stream>>>(part, 256, out_en + layer);
        // expanded energy
        rowstats_kernel<<<2048, 256, 0, stream>>>(Xn, dv, rvec, outN);
        qvec_kernel<<<4, 256, 0, stream>>>(H, rvec, qv);
        // Mm = H^T @ (rv .* Xn)           [1024,outN]
        wmma_gemm_f32<<<dim3((outN + BN - 1) / BN, 1024 / BM), NT, 0, stream>>>(
            H, Xn, Mm, 1024, outN, 2048, 1024, outN, outN, 1, 0, 0, rv);
        edge_energy_kernel<<<1024, 256, 0, stream>>>(Mm, qv, de, perE, outN);
        final_sum_kernel<<<1, 256, 0, stream>>>(perE, 1024, out_enx + layer);

        Xcur = Xn;
    }

    // log_softmax over N (dim=1) of final [2048,40]
    logsoftmax_col_kernel<<<40, 256, 0, stream>>>(Xcur, out_logp, 2048, 40);

    // spectral norms of the 10 weight matrices
    spectral_norm_kernel<<<10, 512, 0, stream>>>(Wf, Wm, Wl, out_sn);

    // spectral gap: power iteration on cI - L^2 -> eigenvalue of L nearest 0
    const float CSHIFT = 4.0f;   // > lambda_max(L)^2 (~2 for this Wigner ensemble)
    sg_init_kernel<<<8, 256, 0, stream>>>(sgv, 2048);
    for (int it = 0; it < 40; ++it) {
        matvec_kernel<<<2048, 256, 0, stream>>>(L, sgv, sgt, 2048);
        matvec_kernel<<<2048, 256, 0, stream>>>(L, sgt, sgu, 2048);
        sg_update_kernel<<<1, 256, 0, stream>>>(sgv, sgu, 2048, CSHIFT);
    }
    matvec_kernel<<<2048, 256, 0, stream>>>(L, sgv, sgt, 2048);
    matvec_kernel<<<2048, 256, 0, stream>>>(L, sgt, sgu, 2048);
    sg_final_kernel<<<1, 256, 0, stream>>>(sgv, sgu, 2048, out_gap);
}